// DeepAttnMIL_Surv_20959440405058
// MI455X (gfx1250) — compile-verified
//
#include <hip/hip_runtime.h>

typedef __attribute__((ext_vector_type(2))) float v2f;
typedef __attribute__((ext_vector_type(8))) float v8f;

constexpr int kB = 8;      // batch
constexpr int kN = 4096;   // patches per batch
constexpr int kD = 1024;   // input feature dim
constexpr int kE = 64;     // embedding dim
constexpr int kC = 10;     // clusters

// ---------------------------------------------------------------------------
// Workspace zeroing (harness poisons d_ws with 0xAA before timing)
// ---------------------------------------------------------------------------
__global__ void zero_ws_kernel(float* __restrict__ ws, int n) {
    int i = blockIdx.x * blockDim.x + threadIdx.x;
    if (i < n) ws[i] = 0.0f;
}

// ---------------------------------------------------------------------------
// Per-(batch,cluster) patch counts (tiny)
// ---------------------------------------------------------------------------
__global__ void count_kernel(const int* __restrict__ labels,
                             float* __restrict__ counts) {
    int i = blockIdx.x * blockDim.x + threadIdx.x;   // i in [0, B*N)
    int b = i / kN;
    int c = labels[i];
    atomicAdd(&counts[b * kC + c], 1.0f);
}

// ---------------------------------------------------------------------------
// Fused: e = relu(data @ W1^T + b1) with fp32 WMMA, segment-summed per cluster.
// Each wave: 2 strips of 16 rows x all 64 output cols (8 f32 accum tiles).
// Block: 4 waves = 128 rows; 128 | 4096 so each block sits in one batch.
// ---------------------------------------------------------------------------
__launch_bounds__(128)
__global__ void embed_gemm_kernel(const float* __restrict__ data,
                                  const int*   __restrict__ labels,
                                  const float* __restrict__ W1,
                                  const float* __restrict__ b1,
                                  float*       __restrict__ g_sums) {
    __shared__ float lds_sums[kC * kE];

    const int tid = threadIdx.x;
    for (int i = tid; i < kC * kE; i += 128) lds_sums[i] = 0.0f;
    __syncthreads();

    const int wave = tid >> 5;
    const int lane = tid & 31;
    const int half = lane >> 4;   // 0: lanes 0-15, 1: lanes 16-31
    const int l16  = lane & 15;

    const int rowBlock = blockIdx.x * 128;       // 128 rows per block
    const int m0       = rowBlock + wave * 32;   // this wave: rows m0..m0+31

    // A fragments (16x4 fp32 layout): lane half 0 holds K{k,k+1}, half 1 K{k+2,k+3}
    const float* pa0 = data + (size_t)(m0 +      l16) * kD + 2 * half;
    const float* pa1 = data + (size_t)(m0 + 16 + l16) * kD + 2 * half;
    // B fragments = W1^T (4x16 fp32 layout, lane = N): B[k][n] = W1[n][k]
    const float* pb0 = W1 + (size_t)( 0 + l16) * kD + 2 * half;
    const float* pb1 = W1 + (size_t)(16 + l16) * kD + 2 * half;
    const float* pb2 = W1 + (size_t)(32 + l16) * kD + 2 * half;
    const float* pb3 = W1 + (size_t)(48 + l16) * kD + 2 * half;

    v8f acc[2][4] = {};   // [strip][ntile], zero-initialized

#pragma unroll 2
    for (int k = 0; k < kD; k += 4) {
        v2f a0 = *(const v2f*)(pa0 + k);
        v2f a1 = *(const v2f*)(pa1 + k);
        v2f b0 = *(const v2f*)(pb0 + k);
        v2f b1f = *(const v2f*)(pb1 + k);
        v2f b2 = *(const v2f*)(pb2 + k);
        v2f b3 = *(const v2f*)(pb3 + k);

        acc[0][0] = __builtin_amdgcn_wmma_f32_16x16x4_f32(false, a0, false, b0,  (short)0, acc[0][0], false, false);
        acc[0][1] = __builtin_amdgcn_wmma_f32_16x16x4_f32(false, a0, false, b1f, (short)0, acc[0][1], false, false);
        acc[0][2] = __builtin_amdgcn_wmma_f32_16x16x4_f32(false, a0, false, b2,  (short)0, acc[0][2], false, false);
        acc[0][3] = __builtin_amdgcn_wmma_f32_16x16x4_f32(false, a0, false, b3,  (short)0, acc[0][3], false, false);
        acc[1][0] = __builtin_amdgcn_wmma_f32_16x16x4_f32(false, a1, false, b0,  (short)0, acc[1][0], false, false);
        acc[1][1] = __builtin_amdgcn_wmma_f32_16x16x4_f32(false, a1, false, b1f, (short)0, acc[1][1], false, false);
        acc[1][2] = __builtin_amdgcn_wmma_f32_16x16x4_f32(false, a1, false, b2,  (short)0, acc[1][2], false, false);
        acc[1][3] = __builtin_amdgcn_wmma_f32_16x16x4_f32(false, a1, false, b3,  (short)0, acc[1][3], false, false);
    }

    // C/D 16x16 f32 layout: VGPR r, lanes 0-15 -> M=r, lanes 16-31 -> M=r+8; N = l16
    int cl[2][8];
#pragma unroll
    for (int s = 0; s < 2; ++s)
#pragma unroll
        for (int r = 0; r < 8; ++r)
            cl[s][r] = labels[m0 + s * 16 + 8 * half + r];

#pragma unroll
    for (int s = 0; s < 2; ++s) {
#pragma unroll
        for (int t = 0; t < 4; ++t) {
            const float bias = b1[t * 16 + l16];
            v8f v = acc[s][t];
#pragma unroll
            for (int r = 0; r < 8; ++r) {
                float e = v[r] + bias;
                e = e > 0.0f ? e : 0.0f;               // ReLU
                atomicAdd(&lds_sums[cl[s][r] * kE + t * 16 + l16], e);  // ds_add_f32
            }
        }
    }
    __syncthreads();

    const int b = rowBlock / kN;
    for (int i = tid; i < kC * kE; i += 128)
        atomicAdd(&g_sums[b * kC * kE + i], lds_sums[i]);
}

// ---------------------------------------------------------------------------
// Head: cluster means, attention MLP, masked softmax, fc6. One block per batch.
// ---------------------------------------------------------------------------
__global__ void head_kernel(const float* __restrict__ g_sums,
                            const float* __restrict__ g_counts,
                            const float* __restrict__ Wa1, const float* __restrict__ ba1,
                            const float* __restrict__ Wa2, const float* __restrict__ ba2,
                            const float* __restrict__ Wf1, const float* __restrict__ bf1,
                            const float* __restrict__ Wf2, const float* __restrict__ bf2,
                            float* __restrict__ out) {
    const int b   = blockIdx.x;
    const int tid = threadIdx.x;  // 0..63

    __shared__ float h[kC][kE];
    __shared__ float mask[kC];
    __shared__ float tanhu[kC][32];
    __shared__ float ascore[kC];
    __shared__ float attn[kC];
    __shared__ float Msh[kE];
    __shared__ float rsh[32];

    for (int c = 0; c < kC; ++c) {
        float cnt = g_counts[b * kC + c];
        h[c][tid] = g_sums[(b * kC + c) * kE + tid] / fmaxf(cnt, 1.0f);
    }
    if (tid < kC) mask[tid] = (g_counts[b * kC + tid] > 0.0f) ? 1.0f : 0.0f;
    __syncthreads();

    // u = tanh(h @ Wa1^T + ba1): threads 0..31 each own one hidden unit
    if (tid < 32) {
        for (int c = 0; c < kC; ++c) {
            float u = ba1[tid];
            for (int e = 0; e < kE; ++e) u += h[c][e] * Wa1[tid * kE + e];
            tanhu[c][tid] = tanhf(u);
        }
    }
    __syncthreads();

    // a[c] = tanhu[c] . Wa2 + ba2
    if (tid < kC) {
        float s = ba2[0];
        for (int j = 0; j < 32; ++j) s += tanhu[tid][j] * Wa2[j];
        ascore[tid] = s;
    }
    __syncthreads();

    // masked softmax, exactly as the reference's numeric trick
    if (tid == 0) {
        float xmax = -__builtin_huge_valf();
        for (int c = 0; c < kC; ++c) {
            float xm = ascore[c] * mask[c] + (1.0f - 1.0f / (mask[c] + 1e-5f));
            xmax = fmaxf(xmax, xm);
        }
        float ex[kC];
        float ssum = 0.0f;
        for (int c = 0; c < kC; ++c) {
            ex[c] = expf(ascore[c] - xmax) * mask[c];
            ssum += ex[c];
        }
        for (int c = 0; c < kC; ++c) attn[c] = ex[c] / ssum;
    }
    __syncthreads();

    // M[e] = sum_c attn[c] * h[c][e]
    {
        float m = 0.0f;
        for (int c = 0; c < kC; ++c) m += attn[c] * h[c][tid];
        Msh[tid] = m;
    }
    __syncthreads();

    // r = relu(M @ Wf1^T + bf1)
    if (tid < 32) {
        float v = bf1[tid];
        for (int e = 0; e < kE; ++e) v += Msh[e] * Wf1[tid * kE + e];
        rsh[tid] = fmaxf(v, 0.0f);
    }
    __syncthreads();

    if (tid == 0) {
        float o = bf2[0];
        for (int j = 0; j < 32; ++j) o += rsh[j] * Wf2[j];
        out[b] = o;
    }
}

// ---------------------------------------------------------------------------
extern "C" void kernel_launch(void* const* d_in, const int* in_sizes, int n_in,
                              void* d_out, int out_size, void* d_ws, size_t ws_size,
                              hipStream_t stream) {
    const float* data   = (const float*)d_in[0];
    const int*   labels = (const int*)  d_in[1];
    const float* W1     = (const float*)d_in[2];
    const float* b1     = (const float*)d_in[3];
    const float* Wa1    = (const float*)d_in[4];
    const float* ba1    = (const float*)d_in[5];
    const float* Wa2    = (const float*)d_in[6];
    const float* ba2    = (const float*)d_in[7];
    const float* Wf1    = (const float*)d_in[8];
    const float* bf1    = (const float*)d_in[9];
    const float* Wf2    = (const float*)d_in[10];
    const float* bf2    = (const float*)d_in[11];
    float* out = (float*)d_out;

    float* g_sums   = (float*)d_ws;                 // [B*C*EMB]
    float* g_counts = g_sums + kB * kC * kE;        // [B*C]

    const int nzero = kB * kC * kE + kB * kC;
    zero_ws_kernel<<<(nzero + 255) / 256, 256, 0, stream>>>(g_sums, nzero);

    count_kernel<<<(kB * kN) / 256, 256, 0, stream>>>(labels, g_counts);

    embed_gemm_kernel<<<(kB * kN) / 128, 128, 0, stream>>>(data, labels, W1, b1, g_sums);

    head_kernel<<<kB, 64, 0, stream>>>(g_sums, g_counts,
                                       Wa1, ba1, Wa2, ba2,
                                       Wf1, bf1, Wf2, bf2, out);
}